// ICPChamferLoss_31696858644903
// MI455X (gfx1250) — compile-verified
//
#include <hip/hip_runtime.h>

typedef __attribute__((ext_vector_type(2))) float v2f;
typedef __attribute__((ext_vector_type(4))) float v4f;
typedef __attribute__((ext_vector_type(8))) float v8f;

#define NPTS            16384
#define TILES           (NPTS / 16)               // 1024 16-row tiles per cloud
#define TILES_PER_WAVE  2
#define BLOCK           256
#define WAVES_PER_BLOCK (BLOCK / 32)              // 8 wave32s
#define TILES_PER_BLOCK (WAVES_PER_BLOCK * TILES_PER_WAVE)   // 16
#define BLOCKS_PER_DIR  (TILES / TILES_PER_BLOCK)            // 64
#define TOTAL_BLOCKS    (2 * BLOCKS_PER_DIR)                 // 128
#define TOTAL_WAVES     (TOTAL_BLOCKS * WAVES_PER_BLOCK)     // 1024

#define CHUNK_TILES     64                        // 64 tiles * 256B = 16KB LDS
#define CHUNK_PTS       (CHUNK_TILES * 16)        // 1024 points per chunk
#define NCHUNKS         (TILES / CHUNK_TILES)     // 16

// ---------------------------------------------------------------------------
// Kernel 1: build augmented 4-component point representations.
//   A-form row:  (-2x, -2y, -2z, 1)        -> WMMA A operand
//   B-form row:  ( x,   y,   z, |p|^2)     -> WMMA B operand
// so (A_src x B_tgt)[i][j] = -2*s_i.t_j + |t_j|^2  (|s_i|^2 added globally).
// ---------------------------------------------------------------------------
__global__ __launch_bounds__(BLOCK)
void chamfer_prep(const float* __restrict__ pred, const float* __restrict__ tgt,
                  v4f* __restrict__ AP, v4f* __restrict__ BP,
                  v4f* __restrict__ AT, v4f* __restrict__ BT) {
  int i = blockIdx.x * BLOCK + threadIdx.x;
  if (i >= NPTS) return;
  float px = pred[3 * i + 0], py = pred[3 * i + 1], pz = pred[3 * i + 2];
  float tx = tgt [3 * i + 0], ty = tgt [3 * i + 1], tz = tgt [3 * i + 2];
  AP[i] = (v4f){-2.f * px, -2.f * py, -2.f * pz, 1.f};
  BP[i] = (v4f){px, py, pz, px * px + py * py + pz * pz};
  AT[i] = (v4f){-2.f * tx, -2.f * ty, -2.f * tz, 1.f};
  BT[i] = (v4f){tx, ty, tz, tx * tx + ty * ty + tz * tz};
}

// ---------------------------------------------------------------------------
// Kernel 2: block stages the target B-array through LDS in 16KB chunks
// (loaded once per block instead of once per wave -> 8x less L2 traffic);
// each wave holds 2 source-tile A frags and runs V_WMMA_F32_16X16X4_F32
// against every target tile, keeping an elementwise running min.
// ---------------------------------------------------------------------------
__global__ __launch_bounds__(BLOCK)
void chamfer_wmma(const float* __restrict__ AP, const float* __restrict__ AT,
                  const v4f* __restrict__ BP4, const v4f* __restrict__ BT4,
                  float* __restrict__ partials) {
  __shared__ v4f lbuf[CHUNK_PTS];                 // 16 KB

  const int wib  = threadIdx.x >> 5;              // wave in block
  const int lane = threadIdx.x & 31;
  const int dir  = blockIdx.x >> 6;               // 0: pred->tgt, 1: tgt->pred
  const int sb   = blockIdx.x & (BLOCKS_PER_DIR - 1);

  const float* __restrict__ Asrc = dir ? AT : AP;
  const v4f*   __restrict__ Btg  = dir ? BP4 : BT4;

  // ISA 7.12.2 fp32 16x4 A layout: lane l -> row l%16, K components
  // 2*(l/16), 2*(l/16)+1 in the two frag VGPRs.  B (4x16) mirrored: N=l%16.
  const int lrow  = lane & 15;
  const int comp0 = (lane >> 4) << 1;

  const int base0 = (sb * TILES_PER_BLOCK + wib * TILES_PER_WAVE) * 16;
  const int base1 = base0 + 16;

  v2f a0 = *(const v2f*)(Asrc + (size_t)(base0 + lrow) * 4 + comp0);
  v2f a1 = *(const v2f*)(Asrc + (size_t)(base1 + lrow) * 4 + comp0);

  const float INF = 3.0e38f;
  v8f m0, m1;
#pragma unroll
  for (int r = 0; r < 8; ++r) { m0[r] = INF; m1[r] = INF; }

  const v8f cz = {};                              // C = 0 each tile
  // Per-lane byte offset of the B fragment inside a 256B tile in LDS:
  // lrow*16 + comp0*4 -> lanes hit all 64 banks exactly once (conflict-free).
  const char* lbase = (const char*)lbuf + lrow * 16 + comp0 * 4;

  for (int c = 0; c < NCHUNKS; ++c) {
    // Cooperative chunk load: 1024 points = 16KB, 4 x b128 per thread.
    __syncthreads();
#pragma unroll
    for (int k = 0; k < CHUNK_PTS / BLOCK; ++k) {
      int idx = threadIdx.x + k * BLOCK;
      lbuf[idx] = Btg[(size_t)c * CHUNK_PTS + idx];
    }
    __syncthreads();

#pragma unroll 4
    for (int t = 0; t < CHUNK_TILES; ++t) {
      v2f b = *(const v2f*)(lbase + t * 256);     // ds_load_b64

      v8f d0 = __builtin_amdgcn_wmma_f32_16x16x4_f32(
          false, a0, false, b, (short)0, cz, false, false);
      v8f d1 = __builtin_amdgcn_wmma_f32_16x16x4_f32(
          false, a1, false, b, (short)0, cz, false, false);

#pragma unroll
      for (int r = 0; r < 8; ++r) {
        m0[r] = __builtin_fminf(m0[r], d0[r]);
        m1[r] = __builtin_fminf(m1[r], d1[r]);
      }
    }
  }

  // Column-min across the 16 lanes of each half-wave (butterfly), then
  // deterministic per-wave partial sum of the 32 row-mins.
  float t0 = 0.f, t1 = 0.f;
#pragma unroll
  for (int r = 0; r < 8; ++r) {
    float v0 = m0[r], v1 = m1[r];
    v0 = __builtin_fminf(v0, __shfl_xor(v0, 8, 16));
    v0 = __builtin_fminf(v0, __shfl_xor(v0, 4, 16));
    v0 = __builtin_fminf(v0, __shfl_xor(v0, 2, 16));
    v0 = __builtin_fminf(v0, __shfl_xor(v0, 1, 16));
    v1 = __builtin_fminf(v1, __shfl_xor(v1, 8, 16));
    v1 = __builtin_fminf(v1, __shfl_xor(v1, 4, 16));
    v1 = __builtin_fminf(v1, __shfl_xor(v1, 2, 16));
    v1 = __builtin_fminf(v1, __shfl_xor(v1, 1, 16));
    t0 += v0;                                     // lanes 0-15: rows base+0..7
    t1 += v1;                                     // lanes 16-31: rows base+8..15
  }
  float tot = t0 + t1;
  tot += __shfl_xor(tot, 16, 32);                 // combine the two half-waves
  if (lane == 0) partials[blockIdx.x * WAVES_PER_BLOCK + wib] = tot;
}

// ---------------------------------------------------------------------------
// Kernel 3: fixed-order reduction of wave partials + global sum of |p|^2 and
// |t|^2 (the per-row constants deferred out of the min), final scaling.
//   out = (sum_partials + sum|p|^2 + sum|t|^2) / (2 * 16384)
// ---------------------------------------------------------------------------
__global__ __launch_bounds__(BLOCK)
void chamfer_reduce(const float* __restrict__ partials,
                    const float* __restrict__ BP, const float* __restrict__ BT,
                    float* __restrict__ out) {
  __shared__ float sm[BLOCK];
  const int t = threadIdx.x;
  float s = 0.f;
  for (int i = t; i < TOTAL_WAVES; i += BLOCK) s += partials[i];
  for (int i = t; i < NPTS; i += BLOCK) s += BP[(size_t)i * 4 + 3];
  for (int i = t; i < NPTS; i += BLOCK) s += BT[(size_t)i * 4 + 3];
  sm[t] = s;
  __syncthreads();
  for (int off = BLOCK / 2; off > 0; off >>= 1) {
    if (t < off) sm[t] += sm[t + off];
    __syncthreads();
  }
  if (t == 0) out[0] = sm[0] * (1.0f / (2.0f * (float)NPTS));
}

// ---------------------------------------------------------------------------
extern "C" void kernel_launch(void* const* d_in, const int* in_sizes, int n_in,
                              void* d_out, int out_size, void* d_ws, size_t ws_size,
                              hipStream_t stream) {
  (void)in_sizes; (void)n_in; (void)out_size; (void)ws_size;
  const float* pred = (const float*)d_in[0];
  const float* tgt  = (const float*)d_in[1];

  float* ws = (float*)d_ws;                 // needs ~1.01 MB of scratch
  float* AP = ws;                           // 16384 * 4 floats each
  float* BP = AP + (size_t)NPTS * 4;
  float* AT = BP + (size_t)NPTS * 4;
  float* BT = AT + (size_t)NPTS * 4;
  float* partials = BT + (size_t)NPTS * 4;  // 1024 floats

  chamfer_prep<<<NPTS / BLOCK, BLOCK, 0, stream>>>(
      pred, tgt, (v4f*)AP, (v4f*)BP, (v4f*)AT, (v4f*)BT);

  chamfer_wmma<<<TOTAL_BLOCKS, BLOCK, 0, stream>>>(
      AP, AT, (const v4f*)BP, (const v4f*)BT, partials);

  chamfer_reduce<<<1, BLOCK, 0, stream>>>(partials, BP, BT, (float*)d_out);
}